// TaskAlignedAssigner_49890340111049
// MI455X (gfx1250) — compile-verified
//
#include <hip/hip_runtime.h>
#include <hip/hip_bf16.h>
#include <stdint.h>

#define TOPK 13
#define NA   8400
#define BS   64
#define NC   80
#define EPSF 1e-9f

typedef float v2f __attribute__((ext_vector_type(2)));
typedef float v8f __attribute__((ext_vector_type(8)));

// ---- gfx1250 async memory helpers (CDNA5 ISA) ----------------

__device__ __forceinline__ uint32_t lds_addr_of(const void* p) {
    // low 32 bits of a generic shared pointer == LDS byte offset on gfx1250
    return (uint32_t)(uintptr_t)p;
}

// GLOBAL_LOAD_ASYNC_TO_LDS_B128 in GVS mode: mem = SGPR64 + VGPR_u32_offset
__device__ __forceinline__ void async_copy_b128(uint32_t lds_addr,
                                                const void* sbase,
                                                uint32_t voff) {
    asm volatile("global_load_async_to_lds_b128 %0, %1, %2"
                 :
                 : "v"(lds_addr), "v"(voff), "s"(sbase)
                 : "memory");
}

__device__ __forceinline__ void wait_async0() {
    asm volatile("s_wait_asynccnt 0" ::: "memory");
}

// ---- kernel A: zero the cross-batch norm accumulator ----------------------

__global__ void zero_norm_kernel(unsigned int* __restrict__ norm_u) {
    int i = blockIdx.x * 256 + threadIdx.x;
    if (i < NA) norm_u[i] = 0u;
}

// ---- kernel B: per-batch IoU + metric + top-13 + mask_pos + norm ---------

__global__ __launch_bounds__(256) void assign_kernel(
    const float* __restrict__ pd_scores,   // (BS, NC, NA)
    const float* __restrict__ pd_bboxes,   // (BS, 4, NA)
    const int*   __restrict__ gt_labels,   // (BS,)
    const float* __restrict__ gt_bboxes,   // (BS, 4)
    float*       __restrict__ out_mask,    // (BS, NA)
    unsigned int* __restrict__ norm_u)     // (NA,) float bits, zero-initialized
{
    __shared__ __align__(16) float s_score[NA];
    __shared__ float s_overlap[NA];
    __shared__ float s_metric[NA];
    __shared__ float r_val[256];
    __shared__ int   r_idx[256];
    __shared__ float s_selval[TOPK];
    __shared__ int   s_selidx[TOPK];
    __shared__ float s_scale;
    __shared__ int   s_mask;

    const int b   = blockIdx.x;
    const int tid = threadIdx.x;

    const int label = gt_labels[b];

    // --- stage the gathered score row into LDS asynchronously (ASYNCcnt) ---
    {
        const uint32_t lds_base = lds_addr_of(&s_score[0]);
        const uint32_t row_off  = ((uint32_t)b * NC + (uint32_t)label) * (NA * 4u);
        for (int t = tid; t < NA / 4; t += 256) {
            async_copy_b128(lds_base + (uint32_t)t * 16u,
                            (const void*)pd_scores,
                            row_off + (uint32_t)t * 16u);
        }
    }

    // --- GT box / area (uniform) ---
    const float gx1 = gt_bboxes[b * 4 + 0], gy1 = gt_bboxes[b * 4 + 1];
    const float gx2 = gt_bboxes[b * 4 + 2], gy2 = gt_bboxes[b * 4 + 3];
    const float area1 = fmaxf(gx2 - gx1, 0.f) * fmaxf(gy2 - gy1, 0.f);

    const float* px1 = pd_bboxes + (size_t)b * 4 * NA;
    const float* py1 = px1 + NA;
    const float* px2 = px1 + 2 * NA;
    const float* py2 = px1 + 3 * NA;

    // --- IoU pass (overlaps with the async score copy) ---
    for (int a = tid; a < NA; a += 256) {
        __builtin_prefetch(&px2[a + 256], 0, 0);   // global_prefetch_b8
        float x1 = px1[a], y1 = py1[a], x2 = px2[a], y2 = py2[a];
        float ow = fmaxf(fminf(gx2, x2) - fmaxf(gx1, x1), 0.f);
        float oh = fmaxf(fminf(gy2, y2) - fmaxf(gy1, y1), 0.f);
        float ov = ow * oh;
        float area2 = fmaxf(x2 - x1, 0.f) * fmaxf(y2 - y1, 0.f);
        s_overlap[a] = ov / (area1 + area2 - ov + EPSF);
    }

    wait_async0();
    __syncthreads();

    // --- align metric = score * iou^6 ---
    for (int a = tid; a < NA; a += 256) {
        float o  = s_overlap[a];
        float o2 = o * o;
        s_metric[a] = s_score[a] * (o2 * o2 * o2);
    }
    __syncthreads();

    // --- iterative top-13 (stable: ties -> smaller index, like lax.top_k) ---
    for (int it = 0; it < TOPK; ++it) {
        float bv = -1.f;
        int   bi = 0x7fffffff;
        for (int a = tid; a < NA; a += 256) {
            float v = s_metric[a];
            if (v > bv) { bv = v; bi = a; }     // ascending a keeps min index on ties
        }
        r_val[tid] = bv; r_idx[tid] = bi;
        __syncthreads();
        for (int s2 = 128; s2 > 0; s2 >>= 1) {
            if (tid < s2) {
                float ov = r_val[tid + s2]; int oi = r_idx[tid + s2];
                if (ov > r_val[tid] || (ov == r_val[tid] && oi < r_idx[tid])) {
                    r_val[tid] = ov; r_idx[tid] = oi;
                }
            }
            __syncthreads();
        }
        if (tid == 0) {
            s_selval[it] = r_val[0];
            s_selidx[it] = r_idx[0];
            s_metric[r_idx[0]] = -1.f;          // remove from further selection
        }
        __syncthreads();
    }

    // --- per-batch scalars ---
    if (tid == 0) {
        int mask = s_selval[0] > EPSF;
        float po = 0.f;
        for (int j = 0; j < TOPK; ++j) po = fmaxf(po, s_overlap[s_selidx[j]]);
        s_scale = po / (s_selval[0] + EPSF);    // pos_overlaps / (pos_align + eps)
        s_mask  = mask;
    }
    __syncthreads();

    // --- mask_pos row (single pass, no global ordering hazards) ---
    const int mask = s_mask;
    float* mrow = out_mask + (size_t)b * NA;
    for (int a = tid; a < NA; a += 256) {
        float m = 0.f;
        if (mask) {
#pragma unroll
            for (int j = 0; j < TOPK; ++j)
                if (a == s_selidx[j]) m = 1.f;
        }
        mrow[a] = m;
    }

    // --- cross-batch max into norm[a] (non-negative floats: uint max == fp max)
    if (mask && tid < TOPK) {
        float c = s_selval[tid] * s_scale;
        atomicMax(&norm_u[s_selidx[tid]], __float_as_uint(c));
    }
}

// ---- kernel C: target_scores = norm (8400) outer gt_labels (64) via WMMA --
// D(16x16) = A(16x4) x B(4x16) with only K=0 populated:
//   A VGPR0, lanes 0-15, K=0  -> norm slice (ISA 7.12.2 "32-bit A 16x4")
//   B VGPR0, lanes 0-15, K=0  -> gt_labels slice (row 0 striped across lanes)
// C/D layout: VGPR j -> M=j (lanes 0-15) / M=j+8 (lanes 16-31), N = lane%16.

__global__ __launch_bounds__(256) void expand_kernel(
    const int*          __restrict__ gt_labels,
    const unsigned int* __restrict__ norm_u,
    float*              __restrict__ out_ts)   // (NA, BS) row-major
{
    const int lane  = threadIdx.x & 31;
    const int wave  = blockIdx.x * 8 + (threadIdx.x >> 5);
    const int ntile = (NA / 16) * (BS / 16);   // 525 * 4 = 2100
    if (wave >= ntile) return;                 // wave-uniform: EXEC stays all-1s

    const int row0 = (wave / (BS / 16)) * 16;  // anchor tile
    const int col0 = (wave % (BS / 16)) * 16;  // batch-column tile

    const int  ln = lane & 15;
    const bool lo = lane < 16;

    float nv = __uint_as_float(norm_u[row0 + ln]);
    float lv = (float)gt_labels[col0 + ln];

    v2f a; a.x = lo ? nv : 0.f; a.y = 0.f;     // K=0 column of A
    v2f bm; bm.x = lo ? lv : 0.f; bm.y = 0.f;  // K=0 row of B
    v8f c = {};

    v8f d = __builtin_amdgcn_wmma_f32_16x16x4_f32(
        /*neg_a=*/false, a, /*neg_b=*/false, bm,
        /*c_mod=*/(short)0, c, /*reuse_a=*/false, /*reuse_b=*/false);

#pragma unroll
    for (int j = 0; j < 8; ++j) {
        int M = j + (lo ? 0 : 8);
        out_ts[(size_t)(row0 + M) * BS + (col0 + ln)] = d[j];
    }
}

// ---------------------------------------------------------------------------

extern "C" void kernel_launch(void* const* d_in, const int* in_sizes, int n_in,
                              void* d_out, int out_size, void* d_ws, size_t ws_size,
                              hipStream_t stream) {
    (void)in_sizes; (void)n_in; (void)out_size; (void)ws_size;

    const float* pd_scores = (const float*)d_in[0];
    const float* pd_bboxes = (const float*)d_in[1];
    const int*   gt_labels = (const int*)d_in[2];
    const float* gt_bboxes = (const float*)d_in[3];

    float* out_ts   = (float*)d_out;                    // target_scores (NA, BS)
    float* out_mask = out_ts + (size_t)NA * BS;         // mask_pos (BS, NA)
    unsigned int* norm_u = (unsigned int*)d_ws;         // NA uints

    zero_norm_kernel<<<(NA + 255) / 256, 256, 0, stream>>>(norm_u);
    assign_kernel<<<BS, 256, 0, stream>>>(pd_scores, pd_bboxes, gt_labels,
                                          gt_bboxes, out_mask, norm_u);
    expand_kernel<<<(2100 + 7) / 8, 256, 0, stream>>>(gt_labels, norm_u, out_ts);
}